// ProjectViewsModel_78640851190456
// MI455X (gfx1250) — compile-verified
//
#include <hip/hip_runtime.h>
#include <hip/hip_bf16.h>
#include <stddef.h>

// ---------------------------------------------------------------------------
// Problem: N=65536 rows, 512 features (emb_a[:,0:256] ++ emb_b[:,0:256]),
// sorted segment_ids into S=4096 segments; segment-mean then [S,512]@[512,32]+b.
// Memory-bound on the 134MB embedding stream; GEMM done with f32 WMMA.
// ---------------------------------------------------------------------------

#define NSEG   4096
#define FEAT   512
#define HALF_F 256
#define ROWS_PER_CHUNK 256

typedef __attribute__((ext_vector_type(2))) float v2f;
typedef __attribute__((ext_vector_type(8))) float v8f;

// --------------------------- kernel 1: zero ws ------------------------------
__global__ void zero_ws_kernel(float* __restrict__ p, size_t n) {
    size_t i = (size_t)blockIdx.x * blockDim.x + threadIdx.x;
    size_t stride = (size_t)gridDim.x * blockDim.x;
    for (; i < n; i += stride) p[i] = 0.0f;
}

// ----------------- kernel 2: sorted segment sum + counts --------------------
// grid  = (2, N/ROWS_PER_CHUNK): x selects emb_a vs emb_b, y selects row chunk
// block = 64 threads; thread t owns columns [4t, 4t+3] of the 256-col source.
// Running sums live in registers; atomics fire only at segment transitions
// (segment_ids is sorted), so atomic traffic is ~#transitions * 512 floats.
__global__ void segsum_kernel(const float* __restrict__ emb_a,
                              const float* __restrict__ emb_b,
                              const int*   __restrict__ seg,
                              float* __restrict__ sums,    // [NSEG, FEAT]
                              float* __restrict__ counts,  // [NSEG]
                              int n_rows) {
    __shared__ int ss[ROWS_PER_CHUNK];
    const int tid  = threadIdx.x;           // 0..63
    const int row0 = blockIdx.y * ROWS_PER_CHUNK;

    for (int i = tid; i < ROWS_PER_CHUNK; i += 64) ss[i] = seg[row0 + i];
    __syncthreads();

    const float* src = (blockIdx.x == 0) ? emb_a : emb_b;
    const int colBase = blockIdx.x * HALF_F;   // 0 or 256 in the 512-wide feature
    const int c4 = tid * 4;                    // column in source (0..252)

    // float4 stream: 16B/thread/row, 512B per wave per row, fully coalesced.
    const float4* p = (const float4*)(src + (size_t)row0 * HALF_F + c4);

    float4 run; run.x = run.y = run.z = run.w = 0.0f;
    int cur = ss[0];

    for (int i = 0; i < ROWS_PER_CHUNK; ++i) {
        int s = ss[i];                          // uniform across block
        if (s != cur) {
            float* d = sums + (size_t)cur * FEAT + colBase + c4;
            atomicAdd(d + 0, run.x);
            atomicAdd(d + 1, run.y);
            atomicAdd(d + 2, run.z);
            atomicAdd(d + 3, run.w);
            run.x = run.y = run.z = run.w = 0.0f;
            cur = s;
        }
        float4 v = p[(size_t)i * (HALF_F / 4)];
        run.x += v.x; run.y += v.y; run.z += v.z; run.w += v.w;
    }
    {
        float* d = sums + (size_t)cur * FEAT + colBase + c4;
        atomicAdd(d + 0, run.x);
        atomicAdd(d + 1, run.y);
        atomicAdd(d + 2, run.z);
        atomicAdd(d + 3, run.w);
    }

    // Row counts: run-length pass over this chunk's (sorted) ids, one thread.
    if (blockIdx.x == 0 && tid == 0) {
        int c = ss[0];
        float r = 0.0f;
        for (int i = 0; i < ROWS_PER_CHUNK; ++i) {
            int s = ss[i];
            if (s != c) { atomicAdd(&counts[c], r); r = 0.0f; c = s; }
            r += 1.0f;
        }
        atomicAdd(&counts[c], r);
    }
    (void)n_rows;
}

// ------------- kernel 3: pooled @ W^T + b via V_WMMA_F32_16X16X4_F32 --------
// out[M=4096, N=32] = (sums[M,512] * inv_count[M]) @ W[32,512]^T + b[32]
// One wave per 16-row M-tile; two accumulators cover N=0..15 and N=16..31.
// A-frag (16x4 f32): M = lane&15 for all lanes; lanes<16 hold K={k,k+1},
// lanes>=16 hold K={k+2,k+3}. B-frag (4x16) mirrors that K split; N = lane&15.
// C/D: VGPR j, M = (lane>>4)*8 + j, N = lane&15.
__global__ void gemm_wmma_kernel(const float* __restrict__ sums,
                                 const float* __restrict__ counts,
                                 const float* __restrict__ W,   // [32, 512] row-major
                                 const float* __restrict__ bias,// [32]
                                 float* __restrict__ out) {     // [NSEG, 32]
    const int wave = threadIdx.x >> 5;
    const int lane = threadIdx.x & 31;
    const int mt   = blockIdx.x * 8 + wave;      // 0..255 M-tiles
    const int m0   = mt * 16;

    const int mrow = m0 + (lane & 15);           // A-matrix row for this lane
    const int n    = lane & 15;                  // B/C column within tile
    const int koff = (lane >> 4) * 2;            // K sub-offset per lane half

    const float cnt = counts[mrow];
    const float inv = 1.0f / fmaxf(cnt, 1.0f);   // fold mean into A load

    const float* arow = sums + (size_t)mrow * FEAT;
    const float* w0   = W + (size_t)n        * FEAT;   // row n       -> N-tile 0
    const float* w1   = W + (size_t)(n + 16) * FEAT;   // row n+16    -> N-tile 1

    v8f acc0 = {};
    v8f acc1 = {};

    #pragma unroll 4
    for (int k = 0; k < FEAT; k += 4) {
        v2f a, b0, b1;
        a.x  = arow[k + koff]     * inv;
        a.y  = arow[k + koff + 1] * inv;
        b0.x = w0[k + koff];
        b0.y = w0[k + koff + 1];
        b1.x = w1[k + koff];
        b1.y = w1[k + koff + 1];
        // 8 args: (neg_a, A, neg_b, B, c_mod, C, reuse_a, reuse_b)
        acc0 = __builtin_amdgcn_wmma_f32_16x16x4_f32(false, a, false, b0,
                                                     (short)0, acc0, false, false);
        acc1 = __builtin_amdgcn_wmma_f32_16x16x4_f32(false, a, false, b1,
                                                     (short)0, acc1, false, false);
    }

    const float bn0 = bias[n];
    const float bn1 = bias[n + 16];
    const int mbase = m0 + (lane >> 4) * 8;
    #pragma unroll
    for (int j = 0; j < 8; ++j) {
        const int m = mbase + j;
        out[(size_t)m * 32 + n]      = acc0[j] + bn0;
        out[(size_t)m * 32 + 16 + n] = acc1[j] + bn1;
    }
}

// ------------------------------- launcher -----------------------------------
extern "C" void kernel_launch(void* const* d_in, const int* in_sizes, int n_in,
                              void* d_out, int out_size, void* d_ws, size_t ws_size,
                              hipStream_t stream) {
    const float* emb_a = (const float*)d_in[0];   // [N, 256] f32
    const float* emb_b = (const float*)d_in[1];   // [N, 256] f32
    const int*   seg   = (const int*)  d_in[2];   // [N] int32, sorted
    const float* W     = (const float*)d_in[3];   // [32, 512] f32
    const float* bias  = (const float*)d_in[4];   // [32] f32
    float*       out   = (float*)d_out;           // [4096, 32] f32

    const int N = in_sizes[2];                    // 65536

    // Workspace layout: sums [NSEG*FEAT] then counts [NSEG]
    float* sums   = (float*)d_ws;
    float* counts = sums + (size_t)NSEG * FEAT;
    const size_t ws_floats = (size_t)NSEG * FEAT + NSEG;

    // 1) zero accumulators (every call: we accumulate atomically below)
    zero_ws_kernel<<<1024, 256, 0, stream>>>(sums, ws_floats);

    // 2) sorted segment sum + counts
    dim3 grid2(2, N / ROWS_PER_CHUNK);
    segsum_kernel<<<grid2, 64, 0, stream>>>(emb_a, emb_b, seg, sums, counts, N);

    // 3) WMMA GEMM: 256 M-tiles, 8 waves/block -> 32 blocks of 256 threads
    gemm_wmma_kernel<<<NSEG / 16 / 8, 256, 0, stream>>>(sums, counts, W, bias, out);

    (void)n_in; (void)out_size; (void)ws_size;
}